// TreeEncoder_58875411693653
// MI455X (gfx1250) — compile-verified
//
#include <hip/hip_runtime.h>

typedef __attribute__((ext_vector_type(2))) float v2f;
typedef __attribute__((ext_vector_type(4))) float v4f;
typedef __attribute__((ext_vector_type(8))) float v8f;

#define NNODE 128
#define DDIM  128
#define HDIM  256
#define LDIM  64
#define SLDS  264   // 256 + 8 padding: breaks the 64-bank alignment for row-strided WMMA fragment loads

// ---------------------------------------------------------------------------
// GEMM: Z[128 x 256] = X[128 x K] @ W[K x 256] + bias, via V_WMMA_F32_16X16X4_F32.
// 8 waves; wave w owns N-tile pair {2w, 2w+1} across all 8 M-tiles
// -> 16 live fp32 accumulators (128 VGPRs), W is streamed from L2 exactly once
//    per block, A-fragments come from LDS.
// ---------------------------------------------------------------------------
__device__ __forceinline__ void gemm256(const float* __restrict__ W,
                                        const float* __restrict__ bias,
                                        const float* __restrict__ X,   // LDS, stride SLDS
                                        float* __restrict__ Z,         // LDS, stride SLDS
                                        int K) {
  const int tid  = threadIdx.x;
  const int lane = tid & 31;
  const int wave = tid >> 5;          // 0..7
  const int l16  = lane & 15;
  const int hf   = lane >> 4;         // 0 or 1: K-half selector per ISA A/B layout
  const int colBase = wave * 32 + l16;

  v8f acc[8][2];
#pragma unroll
  for (int mt = 0; mt < 8; ++mt)
#pragma unroll
    for (int e = 0; e < 8; ++e) { acc[mt][0][e] = 0.0f; acc[mt][1][e] = 0.0f; }

  for (int k = 0; k < K; k += 4) {
    const int kr = k + 2 * hf;
    v2f b0, b1v;
    b0.x  = W[kr * HDIM + colBase];
    b0.y  = W[(kr + 1) * HDIM + colBase];
    b1v.x = W[kr * HDIM + colBase + 16];
    b1v.y = W[(kr + 1) * HDIM + colBase + 16];
#pragma unroll
    for (int mt = 0; mt < 8; ++mt) {
      const int row = mt * 16 + l16;
      v2f a;
      a.x = X[row * SLDS + kr];
      a.y = X[row * SLDS + kr + 1];
      acc[mt][0] = __builtin_amdgcn_wmma_f32_16x16x4_f32(
          false, a, false, b0,  (short)0, acc[mt][0], false, false);
      acc[mt][1] = __builtin_amdgcn_wmma_f32_16x16x4_f32(
          false, a, false, b1v, (short)0, acc[mt][1], false, false);
    }
  }

  const float bs0 = bias[colBase];
  const float bs1 = bias[colBase + 16];
#pragma unroll
  for (int mt = 0; mt < 8; ++mt)
#pragma unroll
    for (int r = 0; r < 8; ++r) {
      const int row = mt * 16 + r + 8 * hf;      // C/D layout: VGPR r -> M = r (+8 for high lanes)
      Z[row * SLDS + colBase]      = acc[mt][0][r] + bs0;
      Z[row * SLDS + colBase + 16] = acc[mt][1][r] + bs1;
    }
}

// ---------------------------------------------------------------------------
// Tree aggregation + ReLU: H[v] = relu(invdeg[v] * (Z[v] + [child]Z[parent] + sum_children Z[c]))
// Thread t owns feature column t; per-v loop bounds are uniform across the block.
// ---------------------------------------------------------------------------
__device__ __forceinline__ void aggregate(const float* __restrict__ Z,
                                          float* __restrict__ H,
                                          const int* parent, const int* childF,
                                          const int* csrOff, const int* csrList,
                                          const float* invdeg) {
  const int f = threadIdx.x;   // 0..255
  for (int v = 0; v < NNODE; ++v) {
    float s = Z[v * SLDS + f];
    if (childF[v]) s += Z[parent[v] * SLDS + f];
    const int e0 = csrOff[v], e1 = csrOff[v + 1];
    for (int e = e0; e < e1; ++e) s += Z[csrList[e] * SLDS + f];
    s *= invdeg[v];
    H[v * SLDS + f] = fmaxf(s, 0.0f);
  }
}

// ---------------------------------------------------------------------------
// One workgroup per graph: BFS tree -> GCN1 -> GCN2 -> pool -> heads, all in LDS.
// ---------------------------------------------------------------------------
__global__ __launch_bounds__(256) void tree_encoder_kernel(
    const float* __restrict__ x,   const float* __restrict__ adj,
    const float* __restrict__ W1,  const float* __restrict__ b1,
    const float* __restrict__ W2,  const float* __restrict__ b2,
    const float* __restrict__ Wmu, const float* __restrict__ bmu,
    const float* __restrict__ Wlv, const float* __restrict__ blv,
    float* __restrict__ out, int B) {
  extern __shared__ float smem[];
  float* bufA = smem;                  // 128 x SLDS  (input / h)
  float* bufZ = smem + NNODE * SLDS;   // 128 x SLDS  (gemm output)

  __shared__ unsigned adjbits[NNODE][4];
  __shared__ int   order_[NNODE];
  __shared__ int   parent_[NNODE];
  __shared__ int   key_[NNODE];        // reused as BFS keys, then CSR counts/cursors
  __shared__ int   childF_[NNODE];
  __shared__ int   csrOff_[NNODE + 1];
  __shared__ int   csrList_[NNODE];
  __shared__ float invdeg_[NNODE];
  __shared__ int   activeF_[NNODE];
  __shared__ unsigned visMask_[4];
  __shared__ float g_[HDIM];
  __shared__ int   cnt_, counter_, nActive_, root_;

  const int tid = threadIdx.x;
  const int b   = blockIdx.x;
  const size_t adjBase = (size_t)b * (NNODE * NNODE);
  const float* __restrict__ xb = x + (size_t)b * (NNODE * DDIM);

  // ---- prefetch this graph's x tile (512 cachelines) so the HBM fetch ----
  // ---- overlaps the BFS phase; GEMM1 staging then hits L2/WGP$        ----
#pragma unroll
  for (int k = 0; k < 2; ++k)
    __builtin_prefetch(xb + (size_t)(tid + k * 256) * 32, 0, 0);   // 128 B per step

  // ---- pack adjacency rows into 128-bit masks (wave32 ballot, coalesced) ----
  {
    const int w8 = tid >> 5, lane = tid & 31;
    const int word = w8 & 3, rowBase = (w8 >> 2) * 64;
    for (int r = 0; r < 64; ++r) {
      const int row = rowBase + r;
      const float v = adj[adjBase + row * NNODE + word * 32 + lane];
      const unsigned m = __builtin_amdgcn_ballot_w32(v > 0.5f);
      if (lane == 0) adjbits[row][word] = m;
    }
  }
  __syncthreads();

  if (tid < NNODE) {
    const unsigned any = adjbits[tid][0] | adjbits[tid][1] | adjbits[tid][2] | adjbits[tid][3];
    activeF_[tid] = (any != 0u);
  }
  if (tid < 4) visMask_[tid] = 0u;
  __syncthreads();
  if (tid == 0) {
    int rt = -1, na = 0;
    for (int v = 0; v < NNODE; ++v) { na += activeF_[v]; if (rt < 0 && activeF_[v]) rt = v; }
    root_ = rt; nActive_ = na;
    counter_ = (rt >= 0) ? 1 : 0;
    if (rt >= 0) visMask_[rt >> 5] |= (1u << (rt & 31));
  }
  __syncthreads();
  const int INFORD = 1 << 20;
  if (tid < NNODE) { order_[tid] = (tid == root_) ? 0 : INFORD; parent_[tid] = 0; }
  __syncthreads();

  // ---- FIFO BFS (level-synchronous; parent = min-order neighbor, ties via key) ----
  for (int iter = 0; iter < NNODE - 1; ++iter) {
    if (tid == 0) cnt_ = 0;
    bool newly = false; int myKey = 0x7FFFFFFF, po = INFORD, p = 0;
    if (tid < NNODE) {
      const bool vis = (visMask_[tid >> 5] >> (tid & 31)) & 1u;
      if (!vis) {
#pragma unroll
        for (int w = 0; w < 4; ++w) {
          unsigned m = adjbits[tid][w] & visMask_[w];
          while (m) {
            const int u = (w << 5) + __builtin_ctz(m); m &= m - 1u;
            const int o = order_[u];
            if (o < po) { po = o; p = u; }
          }
        }
        if (po < INFORD) { newly = true; myKey = po * NNODE + tid; }  // (parent order, index)
      }
      key_[tid] = myKey;
    }
    __syncthreads();
    if (tid < NNODE) {  // waves 0..3 each own one 32-bit visited word
      const unsigned nb = __builtin_amdgcn_ballot_w32(newly);
      if ((tid & 31) == 0) visMask_[tid >> 5] |= nb;
    }
    if (newly) {
      int rank = 0;
      for (int j = 0; j < NNODE; ++j) rank += (key_[j] < myKey);
      order_[tid]  = counter_ + rank;
      parent_[tid] = p;
      atomicAdd(&cnt_, 1);
    }
    __syncthreads();
    if (tid == 0) counter_ += cnt_;
    __syncthreads();
    const int c = cnt_;
    __syncthreads();
    if (c == 0) break;
  }

  // ---- child flags, children CSR, inverse degree ----
  if (tid < NNODE) {
    const bool vis = (visMask_[tid >> 5] >> (tid & 31)) & 1u;
    childF_[tid] = (vis && order_[tid] > 0) ? 1 : 0;
    key_[tid] = 0;
  }
  __syncthreads();
  if (tid < NNODE && childF_[tid]) atomicAdd(&key_[parent_[tid]], 1);
  __syncthreads();
  if (tid == 0) {
    csrOff_[0] = 0;
    for (int v = 0; v < NNODE; ++v) csrOff_[v + 1] = csrOff_[v] + key_[v];
  }
  __syncthreads();
  if (tid < NNODE) key_[tid] = csrOff_[tid];   // cursors
  __syncthreads();
  if (tid < NNODE && childF_[tid]) {
    const int pos = atomicAdd(&key_[parent_[tid]], 1);
    csrList_[pos] = tid;
  }
  __syncthreads();
  if (tid < NNODE) {
    const int deg = 1 + childF_[tid] + (csrOff_[tid + 1] - csrOff_[tid]);
    invdeg_[tid] = 1.0f / (float)deg;
  }

  // ---- stage x[b] (128 x 128) into LDS with B128 loads/stores ----
  for (int i = tid * 4; i < NNODE * DDIM; i += 256 * 4) {
    const v4f val = *reinterpret_cast<const v4f*>(xb + i);
    *reinterpret_cast<v4f*>(&bufA[(i >> 7) * SLDS + (i & 127)]) = val;
  }
  __syncthreads();

  // ---- GCN layer 1: z = x@W1+b1 ; h1 = relu(Anorm z) ----
  gemm256(W1, b1, bufA, bufZ, DDIM);
  __syncthreads();
  aggregate(bufZ, bufA, parent_, childF_, csrOff_, csrList_, invdeg_);
  __syncthreads();

  // ---- GCN layer 2 ----
  gemm256(W2, b2, bufA, bufZ, HDIM);
  __syncthreads();
  aggregate(bufZ, bufA, parent_, childF_, csrOff_, csrList_, invdeg_);
  __syncthreads();

  // ---- masked mean pool over active nodes ----
  {
    float s = 0.0f;
    for (int v = 0; v < NNODE; ++v)
      if (activeF_[v]) s += bufA[v * SLDS + tid];
    const float denom = (float)(nActive_ > 0 ? nActive_ : 1);
    g_[tid] = s / denom;
  }
  __syncthreads();

  // ---- heads: mu / logvar ----
  if (tid < 2 * LDIM) {
    const int which = tid >> 6;             // 0 = mu, 1 = logvar
    const int l = tid & 63;
    const float* Wh = which ? Wlv : Wmu;
    const float* bh = which ? blv : bmu;
    float s = bh[l];
    for (int f = 0; f < HDIM; ++f) s += g_[f] * Wh[f * LDIM + l];
    out[(size_t)which * ((size_t)B * LDIM) + (size_t)b * LDIM + l] = s;
  }
}

extern "C" void kernel_launch(void* const* d_in, const int* in_sizes, int n_in,
                              void* d_out, int out_size, void* d_ws, size_t ws_size,
                              hipStream_t stream) {
  (void)n_in; (void)out_size; (void)d_ws; (void)ws_size;
  const float* x   = (const float*)d_in[0];
  const float* adj = (const float*)d_in[1];
  const float* W1  = (const float*)d_in[2];
  const float* b1  = (const float*)d_in[3];
  const float* W2  = (const float*)d_in[4];
  const float* b2  = (const float*)d_in[5];
  const float* Wmu = (const float*)d_in[6];
  const float* bmu = (const float*)d_in[7];
  const float* Wlv = (const float*)d_in[8];
  const float* blv = (const float*)d_in[9];
  float* out = (float*)d_out;

  const int B = in_sizes[0] / (NNODE * DDIM);   // 1024
  const size_t shmem = (size_t)2 * NNODE * SLDS * sizeof(float);  // ~270 KB of the 320 KB WGP LDS

  hipFuncSetAttribute((const void*)tree_encoder_kernel,
                      hipFuncAttributeMaxDynamicSharedMemorySize, (int)shmem);

  tree_encoder_kernel<<<B, 256, shmem, stream>>>(x, adj, W1, b1, W2, b2,
                                                 Wmu, bmu, Wlv, blv, out, B);
}